// DualLossDiscrete_67001489818054
// MI455X (gfx1250) — compile-verified
//
#include <hip/hip_runtime.h>

#define GEO_N_NODES 250000
#define GEO_N_EDGES 8000000
#define GEO_CUTOFF  2.0f
#define GEO_NACC    (GEO_N_NODES * 3)          // 750000
#define GEO_NPAD    ((GEO_NACC + 63) & ~63)    // 750016: multiple of 64 -> no tail
#define GEO_PARTOFF ((GEO_NACC + 255) & ~255)  // 750080: partials start (no overlap)
#define GEO_RBLOCKS 512

typedef float v2f __attribute__((ext_vector_type(2)));
typedef float v8f __attribute__((ext_vector_type(8)));

// ---------------------------------------------------------------------------
// Kernel 1: zero the (padded) node accumulator. d_ws is poisoned 0xAA by the
// harness and never re-poisoned, so clear on every launch. Padding to NPAD
// lets the WMMA reduction run guard-free.
// ---------------------------------------------------------------------------
__global__ void geo_zero(float* __restrict__ p, int n) {
    int i = blockIdx.x * blockDim.x + threadIdx.x;
    if (i < n) p[i] = 0.0f;
}

// ---------------------------------------------------------------------------
// Kernel 2: fused edge pass.
//   diff = node_eq_global - target_pos_global is linear in the per-edge score,
//   so accumulate s = (edge_inv_g - target_d_global) in ONE scatter
//   (6 atomics/edge instead of 12). Streaming arrays (edge_index /
//   edge_length / edge_inv / lem) are loaded non-temporally so the 136MB
//   stream does not evict the ~10MB of hot node-indexed data + accumulator
//   out of the 192MB L2. Masked edges contribute exactly 0 -> early exit.
// ---------------------------------------------------------------------------
__global__ __launch_bounds__(256) void geo_edge_scatter(
    const float* __restrict__ edge_inv_global,   // (8M,1)
    const float* __restrict__ pos_perturbed,     // (250k,3)
    const float* __restrict__ a,                 // (2048,)
    const float* __restrict__ pos,               // (250k,3)
    const float* __restrict__ edge_length,       // (8M,1)
    const int*   __restrict__ edge_index,        // (2,8M) int32
    const int*   __restrict__ node2graph,        // (250k,) int32
    const unsigned char* __restrict__ is_sidechain,    // (250k,) bool
    const unsigned char* __restrict__ local_edge_mask, // (8M,) bool
    float* __restrict__ acc)                     // (250k,3) accumulator
{
    int e = blockIdx.x * blockDim.x + threadIdx.x;
    if (e >= GEO_N_EDGES) return;

    // NT streamed loads
    const int r   = __builtin_nontemporal_load(edge_index + e);
    const int c   = __builtin_nontemporal_load(edge_index + GEO_N_EDGES + e);
    const float el = __builtin_nontemporal_load(edge_length + e);
    const unsigned char lem = __builtin_nontemporal_load(local_edge_mask + e);

    // L2-resident gathers
    const float prx = pos[3 * r + 0], pry = pos[3 * r + 1], prz = pos[3 * r + 2];
    const float pcx = pos[3 * c + 0], pcy = pos[3 * c + 1], pcz = pos[3 * c + 2];
    const float dx = prx - pcx, dy = pry - pcy, dz = prz - pcz;
    const float d_gt = sqrtf(dx * dx + dy * dy + dz * dz);

    const bool train  = (is_sidechain[r] | is_sidechain[c]) != 0;
    const float d_pert = train ? el : d_gt;

    // global_mask = ((d_pert <= CUTOFF) | lem) & ~lem == (d_pert <= CUTOFF) & !lem
    const bool gmask = (d_pert <= GEO_CUTOFF) && (lem == 0);
    if (!gmask) return;   // score == 0 -> skip remaining gathers + all atomics

    const float ae  = a[node2graph[r]];
    const float d_target = (d_gt - d_pert) * sqrtf(ae) * rsqrtf(1.0f - ae);
    const float s   = __builtin_nontemporal_load(edge_inv_global + e) - d_target;
    const float coef = s / el;   // dd_dr uses 1/edge_length unconditionally

    const float vx = coef * (pos_perturbed[3 * r + 0] - pos_perturbed[3 * c + 0]);
    const float vy = coef * (pos_perturbed[3 * r + 1] - pos_perturbed[3 * c + 1]);
    const float vz = coef * (pos_perturbed[3 * r + 2] - pos_perturbed[3 * c + 2]);

    atomicAdd(&acc[3 * r + 0], vx);
    atomicAdd(&acc[3 * r + 1], vy);
    atomicAdd(&acc[3 * r + 2], vz);
    atomicAdd(&acc[3 * c + 0], -vx);
    atomicAdd(&acc[3 * c + 1], -vy);
    atomicAdd(&acc[3 * c + 2], -vz);
}

// ---------------------------------------------------------------------------
// Kernel 3: sum-of-squares reduction via V_WMMA_F32_16X16X4_F32.
//   Per wave/iteration: each lane does ONE coalesced global_load_b64 of a
//   contiguous float2, squares it into the 16x4 A tile (64 floats/wave),
//   B = ones(4x16), C accumulates row sums across iterations. Loop control is
//   scalarized via readfirstlane (wave-uniform) so EXEC stays all-ones and
//   the branch is SALU; npad is a multiple of 64 so there are no tail guards.
//   After the loop: 8-component lane sum + shfl_xor(16) = full 16x16 total,
//   then LDS combine across the block's 8 waves.
// ---------------------------------------------------------------------------
__global__ __launch_bounds__(256) void geo_sq_reduce_wmma(
    const float* __restrict__ acc, float* __restrict__ partials, int npad)
{
    const int lane = threadIdx.x & 31;
    const int wib  = threadIdx.x >> 5;          // wave in block (0..7)
    const int wave = __builtin_amdgcn_readfirstlane(
        blockIdx.x * (blockDim.x >> 5) + wib);  // scalar wave id
    const int nwaves = gridDim.x * (blockDim.x >> 5);
    const int stride = nwaves * 64;             // floats consumed per sweep

    v8f cacc = {};
    v2f ones; ones[0] = 1.0f; ones[1] = 1.0f;   // B = all-ones (4x16 f32)

    const float* base_ptr = acc + lane * 2;     // lane-contiguous float2
    for (int base = wave * 64; base < npad; base += stride) {
        v2f x = *(const v2f*)(base_ptr + base); // one global_load_b64
        v2f av; av[0] = x[0] * x[0]; av[1] = x[1] * x[1];
        // D = A(16x4) * ones(4x16) + C  -> running sum of all 64 values
        cacc = __builtin_amdgcn_wmma_f32_16x16x4_f32(
            false, av, false, ones, (short)0, cacc, false, false);
    }

    float s = cacc[0] + cacc[1] + cacc[2] + cacc[3] +
              cacc[4] + cacc[5] + cacc[6] + cacc[7];   // 8 rows of my column
    s += __shfl_xor(s, 16, 32);                        // remaining 8 rows

    __shared__ float lds[8];
    if (lane == 0) lds[wib] = s;
    __syncthreads();
    if (threadIdx.x == 0) {
        float t = 0.0f;
        #pragma unroll
        for (int i = 0; i < 8; ++i) t += lds[i];
        partials[blockIdx.x] = t;
    }
}

// ---------------------------------------------------------------------------
// Kernel 4: final reduce of block partials + scale: loss = 2*sum/(3*N_NODES)
// ---------------------------------------------------------------------------
__global__ __launch_bounds__(256) void geo_final(
    const float* __restrict__ partials, int nparts, float* __restrict__ out)
{
    __shared__ float lds[256];
    float s = 0.0f;
    for (int i = threadIdx.x; i < nparts; i += blockDim.x) s += partials[i];
    lds[threadIdx.x] = s;
    __syncthreads();
    for (int off = 128; off > 0; off >>= 1) {
        if (threadIdx.x < off) lds[threadIdx.x] += lds[threadIdx.x + off];
        __syncthreads();
    }
    if (threadIdx.x == 0)
        out[0] = 2.0f * lds[0] / (float)GEO_NACC;
}

// ---------------------------------------------------------------------------
// Inputs (setup_inputs order):
//  0 edge_inv_global f32(8M), 1 pos_perturbed f32(750k), 2 a f32(2048),
//  3 pos f32(750k), 4 edge_length f32(8M), 5 edge_index i32(16M),
//  6 node2graph i32(250k), 7 is_sidechain bool(250k), 8 local_edge_mask bool(8M)
// Output: scalar f32 loss.
// ---------------------------------------------------------------------------
extern "C" void kernel_launch(void* const* d_in, const int* in_sizes, int n_in,
                              void* d_out, int out_size, void* d_ws, size_t ws_size,
                              hipStream_t stream) {
    const float* edge_inv = (const float*)d_in[0];
    const float* pp       = (const float*)d_in[1];
    const float* a        = (const float*)d_in[2];
    const float* pos      = (const float*)d_in[3];
    const float* el       = (const float*)d_in[4];
    const int*   eidx     = (const int*)d_in[5];
    const int*   n2g      = (const int*)d_in[6];
    const unsigned char* sidech = (const unsigned char*)d_in[7];
    const unsigned char* lem    = (const unsigned char*)d_in[8];

    float* acc      = (float*)d_ws;             // GEO_NPAD floats (zeroed)
    float* partials = acc + GEO_PARTOFF;        // GEO_RBLOCKS floats

    geo_zero<<<(GEO_NPAD + 255) / 256, 256, 0, stream>>>(acc, GEO_NPAD);

    geo_edge_scatter<<<(GEO_N_EDGES + 255) / 256, 256, 0, stream>>>(
        edge_inv, pp, a, pos, el, eidx, n2g, sidech, lem, acc);

    geo_sq_reduce_wmma<<<GEO_RBLOCKS, 256, 0, stream>>>(acc, partials, GEO_NPAD);

    geo_final<<<1, 256, 0, stream>>>(partials, GEO_RBLOCKS, (float*)d_out);
}